// Net_7524782703039
// MI455X (gfx1250) — compile-verified
//
#include <hip/hip_runtime.h>
#include <cstddef>

// MI455X / gfx1250, wave32. FP32 WMMA 16x16x4 for both GEMMs (problem is
// memory/L2-bound: ~100MB HBM -> ~4us @ 23.3 TB/s; 4.5 GFLOP f32 WMMA lands
// in the same regime, so full f32 precision is free).
//
// Round-3: out_gemm B panel is staged with GLOBAL_LOAD_ASYNC_TO_LDS_B128
// (ASYNCcnt-tracked, VGPR-bypassing) into a u-major LDS layout so each WMMA
// B fragment is a single ds_load_b64. Hot loops contain no predication.

typedef __attribute__((ext_vector_type(2))) float v2f;
typedef __attribute__((ext_vector_type(4))) float v4f;
typedef __attribute__((ext_vector_type(8))) float v8f;

#define BATCH 8192
#define NUSER 1000

// ---------------------------------------------------------------------------
// colterm[u] = sum_k usr_w[u,k] * ubias_w[u,k]
// ---------------------------------------------------------------------------
__global__ void colterm_kernel(const float* __restrict__ usr_w,
                               const float* __restrict__ ubias_w,
                               float* __restrict__ colterm) {
    int u = blockIdx.x * blockDim.x + threadIdx.x;
    if (u >= NUSER) return;
    const float* a = usr_w + (size_t)u * 128;
    const float* b = ubias_w + (size_t)u * 128;
    float s = 0.f;
#pragma unroll 8
    for (int k = 0; k < 128; ++k) s += a[k] * b[k];
    colterm[u] = s;
}

// ---------------------------------------------------------------------------
// fc0_pad: 96x128, row 95 zeroed so p0_gemm needs no guards.
// ---------------------------------------------------------------------------
__global__ void pad_fc0_kernel(const float* __restrict__ fc0_w,
                               float* __restrict__ fc0_pad) {
    int i = blockIdx.x * blockDim.x + threadIdx.x;   // 96*128 = 12288
    int n = i >> 7;
    int k = i & 127;
    fc0_pad[i] = (n < 95) ? fc0_w[(size_t)n * 128 + k] : 0.f;
}

// ---------------------------------------------------------------------------
// Per-row 11-dim category chain. One wave per batch row (8 rows / 256-thread
// block). Assembles P[row][0..32]    = cat1|cat2|cat3
//                   P[row][128..160] = 0 (cbias zero prefix)
//                   P[row][161..255] = cbiasb_w[c2]
// ---------------------------------------------------------------------------
__global__ void cat_chain_kernel(const int* __restrict__ fcat,
                                 const float* __restrict__ cate_w,
                                 const float* __restrict__ cmat_w,
                                 const float* __restrict__ cbias_w,
                                 const float* __restrict__ cbiasb_w,
                                 float* __restrict__ P) {
    __shared__ float sc[8][33];
    int wave = threadIdx.x >> 5;
    int lane = threadIdx.x & 31;
    int row  = blockIdx.x * 8 + wave;          // grid sized exactly: row < BATCH
    int c0 = fcat[row * 3 + 0];
    int c1 = fcat[row * 3 + 1];
    int c2 = fcat[row * 3 + 2];
    float* s = sc[wave];

    if (lane < 11) s[lane] = cate_w[(size_t)c0 * 11 + lane];
    __syncthreads();
    if (lane < 11) {
        float acc = cbias_w[(size_t)c1 * 11 + lane];
        const float* m = cmat_w + (size_t)c1 * 121 + lane * 11;
#pragma unroll
        for (int j = 0; j < 11; ++j) acc += m[j] * s[j];
        s[11 + lane] = acc;
    }
    __syncthreads();
    if (lane < 11) {
        float acc = cbias_w[(size_t)c2 * 11 + lane];
        const float* m = cmat_w + (size_t)c2 * 121 + lane * 11;
#pragma unroll
        for (int j = 0; j < 11; ++j) acc += m[j] * s[11 + j];
        s[22 + lane] = acc;
    }
    __syncthreads();

    float* Pr = P + (size_t)row * 256;
    for (int i = lane; i < 33; i += 32) Pr[i] = s[i];
    for (int i = lane; i < 33; i += 32) Pr[128 + i] = 0.f;
    for (int j = lane; j < 95; j += 32) Pr[161 + j] = cbiasb_w[(size_t)c2 * 95 + j];
}

// ---------------------------------------------------------------------------
// p0 = feat @ fc0_pad.T + fc0_b   (8192 x 96, K=128), WMMA f32 16x16x4.
// One wave per 16x16 tile. 512 M-tiles x 6 N-tiles = 3072 waves.
// A frag (ISA 32-bit 16x4): lanes 0-15 -> K=k0,k0+1 ; lanes 16-31 -> k0+2,k0+3.
// ---------------------------------------------------------------------------
__global__ void p0_gemm_kernel(const float* __restrict__ feat,
                               const float* __restrict__ fc0_pad,
                               const float* __restrict__ fc0_b,
                               float* __restrict__ p0) {
    int wid   = blockIdx.x * 8 + (threadIdx.x >> 5);
    int lane  = threadIdx.x & 31;
    int mtile = wid / 6;
    int ntile = wid - mtile * 6;
    int hi    = (lane >> 4) & 1;
    int l15   = lane & 15;
    int row   = mtile * 16 + l15;
    int col   = ntile * 16 + l15;               // output feature (n < 95 valid)
    int khoff = hi ? 2 : 0;

    const float* Arow = feat + (size_t)row * 128 + khoff;
    const float* Brow = fc0_pad + (size_t)col * 128 + khoff;

    v8f acc = {};
#pragma unroll 4
    for (int k0 = 0; k0 < 128; k0 += 4) {
        v2f a = *(const v2f*)(Arow + k0);
        v2f b = *(const v2f*)(Brow + k0);
        acc = __builtin_amdgcn_wmma_f32_16x16x4_f32(
            false, a, false, b, (short)0, acc, false, false);
    }

    if (col < 95) {
        float bias = fc0_b[col];
#pragma unroll
        for (int r = 0; r < 8; ++r) {
            int m = mtile * 16 + r + (hi ? 8 : 0);
            p0[(size_t)m * 96 + col] = acc[r] + bias;
        }
    }
}

// ---------------------------------------------------------------------------
// itmq[row] = cvismat_w[c2](95x95) @ p0[row] + cvisbias_w[c2]
// Writes P[row][33..127]; rowdot[row] = sum_j itmq_j * cbiasb_w[c2,j]
// (first 33 cbias columns are zero, so this IS sum(pitm*cbias)).
// Block of 128 threads per row; the 50.5MB gathered table lives in 192MB L2.
// ---------------------------------------------------------------------------
__global__ void itmq_kernel(const int* __restrict__ fcat,
                            const float* __restrict__ p0,
                            const float* __restrict__ cvismat_w,
                            const float* __restrict__ cvisbias_w,
                            const float* __restrict__ cbiasb_w,
                            float* __restrict__ P,
                            float* __restrict__ rowdot) {
    __shared__ float sp[95];
    __shared__ float sred[128];
    int t   = threadIdx.x;
    int row = blockIdx.x;
    int c2  = fcat[row * 3 + 2];

    if (t < 95) sp[t] = p0[(size_t)row * 96 + t];
    __syncthreads();

    float rd = 0.f;
    if (t < 95) {
        const float* Mrow = cvismat_w + (size_t)c2 * 9025 + (size_t)t * 95;
        float acc = cvisbias_w[(size_t)c2 * 95 + t];
#pragma unroll 5
        for (int j = 0; j < 95; ++j) acc += Mrow[j] * sp[j];
        P[(size_t)row * 256 + 33 + t] = acc;
        rd = acc * cbiasb_w[(size_t)c2 * 95 + t];
    }
    sred[t] = rd;
    __syncthreads();
#pragma unroll
    for (int s = 64; s > 0; s >>= 1) {
        if (t < s) sred[t] += sred[t + s];
        __syncthreads();
    }
    if (t == 0) rowdot[row] = sred[0];
}

// ---------------------------------------------------------------------------
// out[b][u] = sum_k P[b][k]*W[u][k] + rowdot[b] + colterm[u]
// W[u] = [usr_w[u] | ubias_w[u]] (256-dim).
//
// Block = 8 waves sharing one ngroup (64 users). The 64x256 B panel is staged
// into LDS (u-major, stride 258) with GLOBAL_LOAD_ASYNC_TO_LDS_B128: each
// 16B chunk is contiguous in global AND LDS, 256 threads move 4KB/pass,
// 16 passes, ASYNCcnt-tracked. Each wave then computes a 16x64 strip: per
// K-step 1 global b64 (A, reused 4x) + 4 ds b64 (B) + 4 WMMA, no predication.
// Grid: 64 mblocks x 16 ngroups = 1024 blocks.
// ---------------------------------------------------------------------------
__global__ void out_gemm_kernel(const float* __restrict__ P,
                                const float* __restrict__ usr_w,
                                const float* __restrict__ ubias_w,
                                const float* __restrict__ rowdot,
                                const float* __restrict__ colterm,
                                float* __restrict__ out) {
    __shared__ float sB[64 * 258];               // [u][k], stride 258 (~64.5KB)

    int ngroup = blockIdx.x & 15;
    int mblock = blockIdx.x >> 4;                // 0..63
    int nbase  = ngroup * 64;
    int t      = threadIdx.x;

    // --- async-stage B panel (zero-fill padded users with plain LDS stores) -
    {
        // generic->LDS offset: low 32 bits of the shared-aperture address
        unsigned ldsBase = (unsigned)(size_t)(void*)sB;
#pragma unroll
        for (int i = 0; i < 16; ++i) {
            int chunk = t + 256 * i;             // 0..4095 16B chunks
            int ulab  = chunk >> 6;              // local user 0..63
            int c     = chunk & 63;              // 16B chunk within 256-k row
            int u     = nbase + ulab;
            int loff  = ulab * 1032 + c * 16;    // bytes (1032 = 258*4)
            if (u < NUSER) {
                const float* src = (c < 32)
                    ? (usr_w  + (size_t)u * 128 + (size_t)c * 4)
                    : (ubias_w + (size_t)u * 128 + (size_t)(c - 32) * 4);
                unsigned dst = ldsBase + (unsigned)loff;
                asm volatile("global_load_async_to_lds_b128 %0, %1, off"
                             :: "v"(dst), "v"(src) : "memory");
            } else {
                v4f z = {};
                *(v4f*)((char*)sB + loff) = z;
            }
        }
        asm volatile("s_wait_asynccnt 0x0" ::: "memory");
    }
    __syncthreads();

    int wave  = t >> 5;
    int lane  = t & 31;
    int mtile = mblock * 8 + wave;               // 0..511
    int hi    = (lane >> 4) & 1;
    int l15   = lane & 15;
    int row   = mtile * 16 + l15;
    int khoff = hi ? 2 : 0;

    const float* Arow = P + (size_t)row * 256 + khoff;
    // u-major panel: B frag (k,k+1) for user ul is contiguous -> one b64 read
    const float* b0p = sB + (size_t)(l15 +  0) * 258 + khoff;
    const float* b1p = sB + (size_t)(l15 + 16) * 258 + khoff;
    const float* b2p = sB + (size_t)(l15 + 32) * 258 + khoff;
    const float* b3p = sB + (size_t)(l15 + 48) * 258 + khoff;

    v8f acc0 = {}, acc1 = {}, acc2 = {}, acc3 = {};
#pragma unroll 4
    for (int k0 = 0; k0 < 256; k0 += 4) {
        v2f a  = *(const v2f*)(Arow + k0);
        v2f b0 = *(const v2f*)(b0p + k0);
        v2f b1 = *(const v2f*)(b1p + k0);
        v2f b2 = *(const v2f*)(b2p + k0);
        v2f b3 = *(const v2f*)(b3p + k0);
        acc0 = __builtin_amdgcn_wmma_f32_16x16x4_f32(false, a, false, b0, (short)0, acc0, false, false);
        acc1 = __builtin_amdgcn_wmma_f32_16x16x4_f32(false, a, false, b1, (short)0, acc1, false, false);
        acc2 = __builtin_amdgcn_wmma_f32_16x16x4_f32(false, a, false, b2, (short)0, acc2, false, false);
        acc3 = __builtin_amdgcn_wmma_f32_16x16x4_f32(false, a, false, b3, (short)0, acc3, false, false);
    }

    int u0 = nbase + l15, u1 = u0 + 16, u2 = u0 + 32, u3 = u0 + 48;
    float ct0 = (u0 < NUSER) ? colterm[u0] : 0.f;
    float ct1 = (u1 < NUSER) ? colterm[u1] : 0.f;
    float ct2 = (u2 < NUSER) ? colterm[u2] : 0.f;
    float ct3 = (u3 < NUSER) ? colterm[u3] : 0.f;
#pragma unroll
    for (int r = 0; r < 8; ++r) {
        int m = mtile * 16 + r + (hi ? 8 : 0);
        float rdv = rowdot[m];
        if (u0 < NUSER) out[(size_t)m * NUSER + u0] = acc0[r] + rdv + ct0;
        if (u1 < NUSER) out[(size_t)m * NUSER + u1] = acc1[r] + rdv + ct1;
        if (u2 < NUSER) out[(size_t)m * NUSER + u2] = acc2[r] + rdv + ct2;
        if (u3 < NUSER) out[(size_t)m * NUSER + u3] = acc3[r] + rdv + ct3;
    }
}

// ---------------------------------------------------------------------------
extern "C" void kernel_launch(void* const* d_in, const int* in_sizes, int n_in,
                              void* d_out, int out_size, void* d_ws, size_t ws_size,
                              hipStream_t stream) {
    (void)in_sizes; (void)n_in; (void)out_size; (void)ws_size;

    const float* feat      = (const float*)d_in[0];
    const int*   fcat      = (const int*)  d_in[1];
    // d_in[2] fusr, d_in[3] fitm, d_in[4] flg : unused on flg==1 path
    const float* usr_w     = (const float*)d_in[5];
    const float* cate_w    = (const float*)d_in[6];
    const float* cmat_w    = (const float*)d_in[7];
    const float* cbias_w   = (const float*)d_in[8];
    const float* ubias_w   = (const float*)d_in[9];
    const float* cbiasb_w  = (const float*)d_in[10];
    const float* fc0_w     = (const float*)d_in[11];
    const float* fc0_b     = (const float*)d_in[12];
    const float* cvismat_w = (const float*)d_in[13];
    const float* cvisbias_w= (const float*)d_in[14];
    float* out = (float*)d_out;

    // ws layout (floats): P[8192*256] | p0[8192*96] | rowdot[8192] |
    //                     colterm[1024] | fc0_pad[96*128]
    float* P       = (float*)d_ws;
    float* p0      = P + (size_t)BATCH * 256;
    float* rowdot  = p0 + (size_t)BATCH * 96;
    float* colterm = rowdot + BATCH;
    float* fc0_pad = colterm + 1024;

    colterm_kernel<<<4, 256, 0, stream>>>(usr_w, ubias_w, colterm);
    pad_fc0_kernel<<<48, 256, 0, stream>>>(fc0_w, fc0_pad);
    cat_chain_kernel<<<BATCH / 8, 256, 0, stream>>>(fcat, cate_w, cmat_w,
                                                    cbias_w, cbiasb_w, P);
    p0_gemm_kernel<<<(512 * 6) / 8, 256, 0, stream>>>(feat, fc0_pad, fc0_b, p0);
    itmq_kernel<<<BATCH, 128, 0, stream>>>(fcat, p0, cvismat_w, cvisbias_w,
                                           cbiasb_w, P, rowdot);
    out_gemm_kernel<<<(64 * 16), 256, 0, stream>>>(P, usr_w, ubias_w,
                                                   rowdot, colterm, out);
}